// CCAPNet_72859825209905
// MI455X (gfx1250) — compile-verified
//
#include <hip/hip_runtime.h>
#include <math.h>

// MI455X / gfx1250, wave32. All matrix math through V_WMMA_F32_16X16X4_F32
// (fp32-exact; problem is bandwidth/latency bound at ~9.5 GFLOP total, so the
// low-throughput fp32 WMMA shape costs nothing and avoids fp16 drift across
// the 128-step recurrence).

typedef __attribute__((ext_vector_type(2))) float v2f;
typedef __attribute__((ext_vector_type(4))) float v4f;
typedef __attribute__((ext_vector_type(8))) float v8f;

#define B_SZ 4096
#define T_SZ 128
#define YV1  31   // YV + 1 output classes

__device__ __forceinline__ v8f wmma_f32_k4(v2f a, v2f b, v8f c) {
  // D(16x16,f32) = A(16x4,f32) * B(4x16,f32) + C
  return __builtin_amdgcn_wmma_f32_16x16x4_f32(false, a, false, b,
                                               (short)0, c, false, false);
}

__device__ __forceinline__ float sigmoidf_(float x) {
  return 1.0f / (1.0f + __expf(-x));
}

// ---------------------------------------------------------------------------
// Embedding table norm-clip (tab * min(1, 1/max(||row||,1e-7)))
// ---------------------------------------------------------------------------
__global__ void scale_table_kernel(const float* __restrict__ tab,
                                   float* __restrict__ stab) {
  int v = blockIdx.x * blockDim.x + threadIdx.x;
  if (v <= 100) {
    float a = tab[v*4+0], b = tab[v*4+1], c = tab[v*4+2], d = tab[v*4+3];
    float nrm = sqrtf(a*a + b*b + c*c + d*d);
    float sc  = fminf(1.0f, 1.0f / fmaxf(nrm, 1e-7f));
    stab[v*4+0] = a*sc; stab[v*4+1] = b*sc;
    stab[v*4+2] = c*sc; stab[v*4+3] = d*sc;
  }
}

__global__ void embed_kernel(const int* __restrict__ xtok,
                             const float* __restrict__ stab,
                             float* __restrict__ emb, int n) {
  int i = blockIdx.x * blockDim.x + threadIdx.x;
  if (i < n) {
    int t = xtok[i];
    v4f e = *(const v4f*)(stab + (size_t)t * 4);
    *(v4f*)(emb + (size_t)i * 4) = e;
  }
}

// ---------------------------------------------------------------------------
// Fused GRU layer: input GEMM (x @ Wih^T + bih) + recurrence + activations
// + masking + optional ragged time reversal + optional residual, all in-reg.
// One wave32 per 16-batch-row tile; h lives in WMMA C/D layout, transposed
// to A layout each step through 1KB of padded LDS.
// ---------------------------------------------------------------------------
template <int DIN, bool BACKWARD, bool HAS_RES>
__global__ void __launch_bounds__(32)
gru_kernel(const float* __restrict__ x,        // [B,T,DIN]
           const float* __restrict__ h0,       // [B,16] or nullptr (zeros)
           const int*   __restrict__ lengths,  // [B]
           const float* __restrict__ Wih,      // [48,DIN]
           const float* __restrict__ Whh,      // [48,16]
           const float* __restrict__ bih,      // [48]
           const float* __restrict__ bhh,      // [48]
           const float* __restrict__ res,      // [B,T,16] or nullptr
           float* __restrict__ out,            // [B,T,16]
           float* __restrict__ hT)             // [B,16]
{
  __shared__ float hsh[16][17];                // +1 pad: kill bank conflicts
  const int lane = threadIdx.x;
  const int r    = lane & 15;                  // A-row / B-col / D-col index
  const int half = lane >> 4;                  // lane half selects K pair / M+8
  const int b0   = blockIdx.x * 16;

  const int lenA = lengths[b0 + r];            // per A-row length
  int lenD[8];                                 // per D-row length (M=j+8*half)
  #pragma unroll
  for (int j = 0; j < 8; ++j) lenD[j] = lengths[b0 + j + 8*half];

  // --- preload weight B-fragments (constant over time) -------------------
  // B(4x16) frag: VGPR0 = {K=k0 (lanes0-15) | K=k0+2 (lanes16-31)}, N=lane%16
  v2f whB[3][4];                               // Whh^T, 3 gates x 4 K-chunks
  #pragma unroll
  for (int g = 0; g < 3; ++g)
    #pragma unroll
    for (int c = 0; c < 4; ++c) {
      int n = g*16 + r, k = 4*c + 2*half;
      whB[g][c] = *(const v2f*)(Whh + n*16 + k);
    }
  constexpr int KC = DIN / 4;
  v2f wiB[3][KC];                              // Wih^T
  #pragma unroll
  for (int g = 0; g < 3; ++g)
    #pragma unroll
    for (int c = 0; c < KC; ++c) {
      int n = g*16 + r, k = 4*c + 2*half;
      wiB[g][c] = *(const v2f*)(Wih + n*DIN + k);
    }
  const float br  = bih[r]      + bhh[r];      // r-gate bias (per N=lane%16)
  const float bz  = bih[16 + r] + bhh[16 + r]; // z-gate bias
  const float bxn = bih[32 + r];               // n-gate x-side bias
  const float bhn = bhh[32 + r];               // n-gate h-side bias

  // --- hidden state: D-layout regs + A-layout frags ----------------------
  float hD[8];
  v2f   hA[4];
  #pragma unroll
  for (int j = 0; j < 8; ++j)
    hD[j] = h0 ? h0[(size_t)(b0 + j + 8*half)*16 + r] : 0.0f;
  #pragma unroll
  for (int c = 0; c < 4; ++c) {
    if (h0) hA[c] = *(const v2f*)(h0 + (size_t)(b0 + r)*16 + 4*c + 2*half);
    else    { hA[c][0] = 0.0f; hA[c][1] = 0.0f; }
  }

  for (int s = 0; s < T_SZ; ++s) {
    // ragged reverse: row r reads x[t = len-1-s] while active
    int tA = s;
    if (BACKWARD) tA = (s < lenA) ? (lenA - 1 - s) : s;

    v2f xA[KC];
    #pragma unroll
    for (int c = 0; c < KC; ++c)
      xA[c] = *(const v2f*)(x + ((size_t)(b0 + r)*T_SZ + tA)*DIN
                              + 4*c + 2*half);
    if (s + 1 < T_SZ) {       // hint next step's activations into cache
      int tn = BACKWARD ? ((s + 1 < lenA) ? (lenA - 2 - s) : (s + 1)) : (s + 1);
      __builtin_prefetch(x + ((size_t)(b0 + r)*T_SZ + tn)*DIN, 0, 1);
    }

    v8f ar, az, axn, ahn;                      // gate pre-activations
    #pragma unroll
    for (int j = 0; j < 8; ++j) { ar[j] = br; az[j] = bz;
                                  axn[j] = bxn; ahn[j] = bhn; }

    #pragma unroll
    for (int c = 0; c < KC; ++c) {             // x @ Wih^T (fused input GEMM)
      ar  = wmma_f32_k4(xA[c], wiB[0][c], ar);
      az  = wmma_f32_k4(xA[c], wiB[1][c], az);
      axn = wmma_f32_k4(xA[c], wiB[2][c], axn);
    }
    #pragma unroll
    for (int c = 0; c < 4; ++c) {              // h @ Whh^T (recurrence)
      ar  = wmma_f32_k4(hA[c], whB[0][c], ar);
      az  = wmma_f32_k4(hA[c], whB[1][c], az);
      ahn = wmma_f32_k4(hA[c], whB[2][c], ahn);
    }

    __syncthreads();  // prev-iter hA reads vs. this iter's hsh writes
    #pragma unroll
    for (int j = 0; j < 8; ++j) {
      float rg   = sigmoidf_(ar[j]);
      float zg   = sigmoidf_(az[j]);
      float ng   = tanhf(axn[j] + rg * ahn[j]);
      float hnew = (1.0f - zg) * ng + zg * hD[j];
      bool  act  = s < lenD[j];
      float hm   = act ? hnew : hD[j];         // freeze past length
      float ov   = act ? hnew : 0.0f;          // zero padded outputs
      hD[j] = hm;
      int m  = b0 + j + 8*half;
      int tD = s;
      if (BACKWARD) tD = act ? (lenD[j] - 1 - s) : s;  // un-reverse on store
      size_t oi = ((size_t)m * T_SZ + tD) * 16 + r;    // coalesced per half
      if (HAS_RES) ov += res[oi];              // fused residual add
      out[oi] = ov;
      hsh[j + 8*half][r] = hm;                 // D layout -> LDS
    }
    __syncthreads();
    #pragma unroll
    for (int c = 0; c < 4; ++c) {              // LDS -> A layout
      hA[c][0] = hsh[r][4*c + 2*half];
      hA[c][1] = hsh[r][4*c + 2*half + 1];
    }
  }

  if (hT)
    #pragma unroll
    for (int j = 0; j < 8; ++j)
      hT[(size_t)(b0 + j + 8*half)*16 + r] = hD[j];
}

// ---------------------------------------------------------------------------
// Final projection: concat(hf,hb) [B*T,32] @ W^T [32,31] + b, WMMA-tiled.
// One wave per 16-row tile; N padded 31->32 with a guarded column.
// ---------------------------------------------------------------------------
__global__ void __launch_bounds__(256)
proj_kernel(const float* __restrict__ hf, const float* __restrict__ hb,
            const float* __restrict__ W,  const float* __restrict__ bvec,
            float* __restrict__ outp)
{
  const int lane = threadIdx.x & 31;
  const int wave = threadIdx.x >> 5;
  const int r    = lane & 15;
  const int half = lane >> 4;
  const size_t row0 = ((size_t)blockIdx.x * 8 + wave) * 16;

  v2f wB[2][8];
  v8f acc[2];
  #pragma unroll
  for (int nt = 0; nt < 2; ++nt) {
    int n = nt*16 + r;
    float bv = (n < YV1) ? bvec[n] : 0.0f;
    #pragma unroll
    for (int j = 0; j < 8; ++j) acc[nt][j] = bv;     // bias in accumulator
    #pragma unroll
    for (int c = 0; c < 8; ++c) {
      int k = 4*c + 2*half;
      if (n < YV1) wB[nt][c] = *(const v2f*)(W + (size_t)n*32 + k);
      else         { wB[nt][c][0] = 0.0f; wB[nt][c][1] = 0.0f; }
    }
  }
  v2f aA[8];                                         // K=32 = [hf | hb]
  #pragma unroll
  for (int c = 0; c < 8; ++c) {
    int k = 4*c + 2*half;
    const float* src = (k < 16) ? hf : hb;
    aA[c] = *(const v2f*)(src + (row0 + r)*16 + (k & 15));
  }
  #pragma unroll
  for (int c = 0; c < 8; ++c) {
    acc[0] = wmma_f32_k4(aA[c], wB[0][c], acc[0]);
    acc[1] = wmma_f32_k4(aA[c], wB[1][c], acc[1]);
  }
  #pragma unroll
  for (int nt = 0; nt < 2; ++nt) {
    int n = nt*16 + r;
    if (n < YV1) {
      #pragma unroll
      for (int j = 0; j < 8; ++j)
        outp[(row0 + j + 8*half) * YV1 + n] = acc[nt][j];
    }
  }
}

// ---------------------------------------------------------------------------
extern "C" void kernel_launch(void* const* d_in, const int* in_sizes, int n_in,
                              void* d_out, int out_size, void* d_ws,
                              size_t ws_size, hipStream_t stream)
{
  const int*   x    = (const int*)d_in[0];
  const int*   lens = (const int*)d_in[1];
  const float* tab  = (const float*)d_in[2];
  // params order: bi_f@3, bi_b@7, f1@11, b1@15, f2@19, b2@23 (Wih,Whh,bih,bhh)
  const float* W    = (const float*)d_in[27];
  const float* bv   = (const float*)d_in[28];
  auto L = [&](int i) { return (const float*)d_in[i]; };

  const size_t NT   = (size_t)B_SZ * T_SZ;
  float* ws   = (float*)d_ws;
  float* stab = ws;                   // 512 floats (16B aligned regions)
  float* emb  = stab + 512;           // [B,T,4]
  float* xf0  = emb  + NT * 4;        // [B,T,16]
  float* xb0  = xf0  + NT * 16;
  float* xf1  = xb0  + NT * 16;
  float* xb1  = xf1  + NT * 16;
  float* hf0  = xb1  + NT * 16;       // [B,16] hidden carries
  float* hb0  = hf0  + (size_t)B_SZ * 16;
  float* hf1  = hb0  + (size_t)B_SZ * 16;
  float* hb1  = hf1  + (size_t)B_SZ * 16;
  float* hdmp = hb1  + (size_t)B_SZ * 16;  // dummy hT for last layers

  scale_table_kernel<<<1, 128, 0, stream>>>(tab, stab);
  embed_kernel<<<(int)((NT + 255) / 256), 256, 0, stream>>>(x, stab, emb,
                                                            (int)NT);
  // layer 1: bidirectional over embeddings (only bi_b is time-reversed)
  gru_kernel<4, false, false><<<B_SZ/16, 32, 0, stream>>>(
      emb, nullptr, lens, L(3), L(4), L(5), L(6), nullptr, xf0, hf0);
  gru_kernel<4, true, false><<<B_SZ/16, 32, 0, stream>>>(
      emb, nullptr, lens, L(7), L(8), L(9), L(10), nullptr, xb0, hb0);
  // layer 2 (forward in time for both streams) + fused residual (out = gru+in)
  gru_kernel<16, false, true><<<B_SZ/16, 32, 0, stream>>>(
      xf0, hf0, lens, L(11), L(12), L(13), L(14), xf0, xf1, hf1);
  gru_kernel<16, false, true><<<B_SZ/16, 32, 0, stream>>>(
      xb0, hb0, lens, L(15), L(16), L(17), L(18), xb0, xb1, hb1);
  // layer 3 + residual; outputs reuse xf0/xb0 (dead after layer 2)
  gru_kernel<16, false, true><<<B_SZ/16, 32, 0, stream>>>(
      xf1, hf1, lens, L(19), L(20), L(21), L(22), xf1, xf0, hdmp);
  gru_kernel<16, false, true><<<B_SZ/16, 32, 0, stream>>>(
      xb1, hb1, lens, L(23), L(24), L(25), L(26), xb1, xb0, hdmp);
  // final projection over concat(xf, xb)
  proj_kernel<<<(int)(NT / 16 / 8), 256, 0, stream>>>(xf0, xb0, W, bv,
                                                      (float*)d_out);
}